// GATConv_9663676416812
// MI455X (gfx1250) — compile-verified
//
#include <hip/hip_runtime.h>
#include <hip/hip_bf16.h>
#include <cstdint>

typedef __attribute__((ext_vector_type(2))) float v2f;
typedef __attribute__((ext_vector_type(8))) float v8f;

#define HD        128   // H*D output features (also F input features)
#define NHEAD     4
#define DHEAD     32
#define NEG_SLOPE 0.2f

// ---------------------------------------------------------------------------
// Kernel 1: proj = feat @ W^T  via V_WMMA_F32_16X16X4_F32 (exact fp32).
// One wave per 16-row block; 8 column tiles of 16; K=128 in steps of 4.
// A (16x4 f32): lane l -> row M=l&15, vgpr v + 2*(l>>4) = K.
// B (4x16 f32): lane l -> col N=l&15, vgpr v + 2*(l>>4) = K;  B[k][n] = W[n][k].
// D (16x16 f32): vgpr v, lane l -> row M = v + 8*(l>>4), col N = l&15.
// ---------------------------------------------------------------------------
__global__ __launch_bounds__(32)
void gat_proj_wmma(const float* __restrict__ feat, const float* __restrict__ W,
                   float* __restrict__ proj, int N)
{
    const int lane    = threadIdx.x & 31;
    const int half    = lane >> 4;
    const int l16     = lane & 15;
    const int rowBase = blockIdx.x * 16;

    int rowA = rowBase + l16;
    if (rowA >= N) rowA = N - 1;          // clamp loads; stores are guarded

    v8f acc[8] = {};                       // 8 n-tiles of 16 cols, f32 C/D

    const float* frow = feat + (size_t)rowA * HD + 2 * half;

    for (int kk = 0; kk < HD; kk += 4) {
        v2f a;
        a.x = frow[kk + 0];                // K = kk + 2*half
        a.y = frow[kk + 1];                // K = kk + 2*half + 1
        #pragma unroll
        for (int t = 0; t < 8; ++t) {
            const float* wrow = W + (size_t)(t * 16 + l16) * HD + kk + 2 * half;
            v2f b;
            b.x = wrow[0];
            b.y = wrow[1];
            acc[t] = __builtin_amdgcn_wmma_f32_16x16x4_f32(
                false, a, false, b, (short)0, acc[t], false, false);
        }
    }

    #pragma unroll
    for (int t = 0; t < 8; ++t) {
        #pragma unroll
        for (int v = 0; v < 8; ++v) {
            const int row = rowBase + v + 8 * half;
            if (row < N)
                proj[(size_t)row * HD + t * 16 + l16] = acc[t][v];
        }
    }
}

// ---------------------------------------------------------------------------
// Kernel 2: per-node attention halves el/er: one thread per (node, head).
// ---------------------------------------------------------------------------
__global__ __launch_bounds__(256)
void gat_el_er(const float* __restrict__ proj, const float* __restrict__ attn_l,
               const float* __restrict__ attn_r, float* __restrict__ el,
               float* __restrict__ er, int N)
{
    const int idx = blockIdx.x * blockDim.x + threadIdx.x;
    if (idx >= N * NHEAD) return;
    const int n = idx >> 2;
    const int h = idx & 3;

    const float* p  = proj   + (size_t)n * HD + h * DHEAD;
    const float* al = attn_l + h * DHEAD;
    const float* ar = attn_r + h * DHEAD;

    float sl = 0.f, sr = 0.f;
    #pragma unroll
    for (int d = 0; d < DHEAD; ++d) {
        const float x = p[d];
        sl += x * al[d];
        sr += x * ar[d];
    }
    el[idx] = sl;
    er[idx] = sr;
}

// ---------------------------------------------------------------------------
// Kernel 3: CSR row offsets from sorted dst via parallel lower_bound.
// row_ptr[n] = first edge with dst >= n;  row_ptr[N] = E.
// ---------------------------------------------------------------------------
__global__ __launch_bounds__(256)
void build_row_ptr(const long long* __restrict__ dst, int* __restrict__ row_ptr,
                   int N, int E)
{
    const int n = blockIdx.x * blockDim.x + threadIdx.x;
    if (n > N) return;
    int lo = 0, hi = E;
    while (lo < hi) {
        const int mid = (lo + hi) >> 1;
        if (dst[mid] < (long long)n) lo = mid + 1; else hi = mid;
    }
    row_ptr[n] = lo;
}

// ---------------------------------------------------------------------------
// Kernel 4: per-dst-node segment softmax + weighted aggregation.
// One wave32 per destination node (mean degree ~16).
// Phase 1: leaky-relu scores -> ex[], wave max per head.
// Phase 2: exp(e - m) -> ex[], wave sum per head.
// Phase 3: whole wave walks the segment; lane owns 4 contiguous dims
//          (coalesced 512B row read of proj[src] per edge).
// ---------------------------------------------------------------------------
__global__ __launch_bounds__(256)
void gat_aggregate(const float* __restrict__ proj, const float* __restrict__ el,
                   const float* __restrict__ er, const long long* __restrict__ src,
                   const int* __restrict__ row_ptr, const float* __restrict__ bias,
                   float* __restrict__ ex, float* __restrict__ out, int N)
{
    const int lane = threadIdx.x & 31;
    const int node = blockIdx.x * (blockDim.x >> 5) + (threadIdx.x >> 5);
    if (node >= N) return;

    const int es = row_ptr[node];
    const int ee = row_ptr[node + 1];

    float ern[NHEAD];
    #pragma unroll
    for (int h = 0; h < NHEAD; ++h) ern[h] = er[(size_t)node * NHEAD + h];

    // ---- phase 1: scores + segment max -----------------------------------
    float mx[NHEAD];
    #pragma unroll
    for (int h = 0; h < NHEAD; ++h) mx[h] = -1e30f;

    for (int i = es + lane; i < ee; i += 32) {
        const long long s = src[i];
        #pragma unroll
        for (int h = 0; h < NHEAD; ++h) {
            float e = el[(size_t)s * NHEAD + h] + ern[h];
            e = (e >= 0.f) ? e : NEG_SLOPE * e;
            ex[(size_t)i * NHEAD + h] = e;
            mx[h] = fmaxf(mx[h], e);
        }
    }
    #pragma unroll
    for (int off = 16; off > 0; off >>= 1) {
        #pragma unroll
        for (int h = 0; h < NHEAD; ++h)
            mx[h] = fmaxf(mx[h], __shfl_xor(mx[h], off, 32));
    }

    // ---- phase 2: exp + segment sum --------------------------------------
    float den[NHEAD] = {0.f, 0.f, 0.f, 0.f};
    for (int i = es + lane; i < ee; i += 32) {
        #pragma unroll
        for (int h = 0; h < NHEAD; ++h) {
            const float v = __expf(ex[(size_t)i * NHEAD + h] - mx[h]);
            ex[(size_t)i * NHEAD + h] = v;
            den[h] += v;
        }
    }
    #pragma unroll
    for (int off = 16; off > 0; off >>= 1) {
        #pragma unroll
        for (int h = 0; h < NHEAD; ++h)
            den[h] += __shfl_xor(den[h], off, 32);
    }

    // ---- phase 3: weighted aggregation -----------------------------------
    const int   hh    = lane >> 3;                  // head owning lane's dims
    const int   dbase = lane * 4;
    const float rden  = (den[hh] > 0.f) ? (1.f / den[hh]) : 0.f;

    float a0 = 0.f, a1 = 0.f, a2 = 0.f, a3 = 0.f;
    for (int i = es; i < ee; ++i) {
        const long long s = src[i];
        const float4 p = *reinterpret_cast<const float4*>(
            proj + (size_t)s * HD + dbase);
        const float  a = ex[(size_t)i * NHEAD + hh] * rden;
        a0 += p.x * a; a1 += p.y * a; a2 += p.z * a; a3 += p.w * a;
    }

    float* o = out + (size_t)node * HD + dbase;
    o[0] = a0 + bias[dbase + 0];
    o[1] = a1 + bias[dbase + 1];
    o[2] = a2 + bias[dbase + 2];
    o[3] = a3 + bias[dbase + 3];
}

// ---------------------------------------------------------------------------
extern "C" void kernel_launch(void* const* d_in, const int* in_sizes, int n_in,
                              void* d_out, int out_size, void* d_ws, size_t ws_size,
                              hipStream_t stream)
{
    const float*     feat   = (const float*)d_in[0];
    const float*     W      = (const float*)d_in[1];
    const float*     attn_l = (const float*)d_in[2];
    const float*     attn_r = (const float*)d_in[3];
    const float*     bias   = (const float*)d_in[4];
    const long long* src    = (const long long*)d_in[5];
    const long long* dst    = (const long long*)d_in[6];

    const int N = in_sizes[0] / HD;   // 100000
    const int E = in_sizes[5];        // 1600000

    // workspace layout
    char*  ws     = (char*)d_ws;
    float* proj   = (float*)ws;  ws += (size_t)N * HD    * sizeof(float);
    float* el     = (float*)ws;  ws += (size_t)N * NHEAD * sizeof(float);
    float* er     = (float*)ws;  ws += (size_t)N * NHEAD * sizeof(float);
    float* ex     = (float*)ws;  ws += (size_t)E * NHEAD * sizeof(float);
    int*   rowptr = (int*)ws;

    gat_proj_wmma<<<(N + 15) / 16, 32, 0, stream>>>(feat, W, proj, N);
    gat_el_er<<<(N * NHEAD + 255) / 256, 256, 0, stream>>>(proj, attn_l, attn_r,
                                                           el, er, N);
    build_row_ptr<<<(N + 1 + 255) / 256, 256, 0, stream>>>(dst, rowptr, N, E);
    gat_aggregate<<<(N + 7) / 8, 256, 0, stream>>>(proj, el, er, src, rowptr,
                                                   bias, ex, (float*)d_out, N);
}